// ProbMaskFilter_22316650070837
// MI455X (gfx1250) — compile-verified
//
#include <hip/hip_runtime.h>

// ProbMask + ThresholdProbMask + UnderSample fusion for MI455X (gfx1250).
// Roofline: 134 MB read + 33.5 MB write @ 23.3 TB/s ~= 7.2 us; 0.35 FLOP/byte
// -> pure bandwidth streaming. CDNA5 path used: async global->LDS (ASYNCcnt)
// with a 3-deep ring buffer for latency hiding, split wait counters, NT stores.

typedef float float4v __attribute__((ext_vector_type(4)));

static constexpr float PM_SLOPE = 5.0f;
static constexpr float S_SLOPE  = 12.0f;

static constexpr int BLOCK  = 256;     // 8 wave32
static constexpr int GRID   = 1024;
static constexpr int PIXC   = 1024;    // pixels per chunk (BLOCK * 4)
static constexpr int CHUNKS = 8;       // chunks per block
static constexpr int NBUF   = 3;       // LDS ring depth (prefetch distance 2)
// GRID * CHUNKS * PIXC == 32*512*512 == 8,388,608 pixels, exact fit (no tails,
// EXEC all-ones everywhere as the async/LDS paths prefer).

// Issue 4 async 128-bit global->LDS loads for this lane (one chunk's worth).
// The memory stride between quarters (256 px * 16B = 4096B) equals the LDS
// stride (256 slots * 16B), and the instruction offset is added to BOTH the
// global and the LDS address (ISA 15.18), so one address pair serves all four.
__device__ __forceinline__ void issue4(unsigned lds_byte_addr, const float4v* src) {
    unsigned long long ga = (unsigned long long)src;
    asm volatile("global_load_async_to_lds_b128 %0, %1, off"
                 :: "v"(lds_byte_addr), "v"(ga) : "memory");
    asm volatile("global_load_async_to_lds_b128 %0, %1, off offset:4096"
                 :: "v"(lds_byte_addr), "v"(ga) : "memory");
    asm volatile("global_load_async_to_lds_b128 %0, %1, off offset:8192"
                 :: "v"(lds_byte_addr), "v"(ga) : "memory");
    asm volatile("global_load_async_to_lds_b128 %0, %1, off offset:12288"
                 :: "v"(lds_byte_addr), "v"(ga) : "memory");
}

// Compile-time-selected wait (loop is fully unrolled; dead branches fold).
__device__ __forceinline__ void wait_async_le(int n) {
    if (n >= 8)      asm volatile("s_wait_asynccnt 8" ::: "memory");
    else if (n == 4) asm volatile("s_wait_asynccnt 4" ::: "memory");
    else             asm volatile("s_wait_asynccnt 0" ::: "memory");
}

__global__ __launch_bounds__(BLOCK) void probmask_undersample_kernel(
    const float4v* __restrict__ x,      // (32,512,512,4) f32, one float4 per pixel
    const float*   __restrict__ w,      // (1,8,8,4) f32
    const float*   __restrict__ thr,    // (1,8,8,4) f32
    float*         __restrict__ out)    // (32,512,512,1) f32
{
    __shared__ float4v s_x[NBUF][PIXC]; // 48 KB ring buffer
    __shared__ float4v s_mask[64];      // [(h%8)*8 + (w%8)] -> 4 channel weights (1 KB)

    const int t = threadIdx.x;

    // --- mask precompute: 256 threads, one scalar each (cost ~0) ---
    {
        float e1 = __expf(-PM_SLOPE * w[t]);
        float pm = __builtin_amdgcn_rcpf(1.0f + e1);              // sigmoid(5*w)
        float e2 = __expf(-S_SLOPE * (pm - thr[t]));
        float m  = __builtin_amdgcn_rcpf(1.0f + e2);              // sigmoid(12*(pm-thr))
        reinterpret_cast<float*>(s_mask)[t] = m;
    }
    __syncthreads();

    const int base = (int)blockIdx.x * (CHUNKS * PIXC);
    const float4v* xp = x + base + t;
    // Byte offset of this lane's buffer-0 staging slot within the workgroup's
    // LDS allocation (flat LDS addresses carry the offset in the low 32 bits).
    const unsigned lds0 = (unsigned)(size_t)&s_x[0][t];

    // Prologue: prefetch chunks 0 and 1 (8 async loads in flight).
    issue4(lds0,           xp);
    issue4(lds0 + 16384u,  xp + PIXC);

    #pragma unroll
    for (int j = 0; j < CHUNKS; ++j) {
        const int buf   = j % NBUF;
        const int pbase = base + j * PIXC;

        if (j + 2 < CHUNKS) {
            // WAR guard: async LDS writes are unordered vs DS reads — ensure
            // iteration j-1's ds_loads from buffer (j+2)%3 retired before
            // the async engine overwrites it.
            asm volatile("s_wait_dscnt 0" ::: "memory");
            issue4(lds0 + (unsigned)((j + 2) % NBUF) * 16384u, xp + (j + 2) * PIXC);
        }
        // Async loads complete in order: <=8 outstanding (2 chunks ahead)
        // means chunk j has fully landed in LDS.
        {
            const int rem = (CHUNKS - 1) - j;          // chunks still in flight
            wait_async_le(rem >= 2 ? 8 : rem * 4);
        }

        // Each chunk covers exactly 2 image rows (1024 px / 512 px-per-row),
        // starting on an even row. 512 % 8 == 0 => row = (p>>9)&7, col = p&7.
        const int row0 = (pbase >> 9) & 7;
        const int col  = t & 7;
        const float4v m0 = s_mask[(row0 << 3) | col];
        const float4v m1 = s_mask[(((row0 + 1) & 7) << 3) | col];

        float4v r0 = s_x[buf][t];
        float4v r1 = s_x[buf][256 + t];
        float4v r2 = s_x[buf][512 + t];
        float4v r3 = s_x[buf][768 + t];

        float o0 = r0.x * m0.x + r0.y * m0.y + r0.z * m0.z + r0.w * m0.w;
        float o1 = r1.x * m0.x + r1.y * m0.y + r1.z * m0.z + r1.w * m0.w;
        float o2 = r2.x * m1.x + r2.y * m1.y + r2.z * m1.z + r2.w * m1.w;
        float o3 = r3.x * m1.x + r3.y * m1.y + r3.z * m1.z + r3.w * m1.w;

        float* o = out + pbase + t;
        __builtin_nontemporal_store(o0, o);
        __builtin_nontemporal_store(o1, o + 256);
        __builtin_nontemporal_store(o2, o + 512);
        __builtin_nontemporal_store(o3, o + 768);
    }
}

extern "C" void kernel_launch(void* const* d_in, const int* in_sizes, int n_in,
                              void* d_out, int out_size, void* d_ws, size_t ws_size,
                              hipStream_t stream) {
    (void)in_sizes; (void)n_in; (void)out_size; (void)d_ws; (void)ws_size;
    const float4v* x   = (const float4v*)d_in[0];   // (32,512,512,4) f32
    const float*   w   = (const float*)d_in[1];     // (1,8,8,4) f32
    const float*   thr = (const float*)d_in[2];     // (1,8,8,4) f32
    float*         out = (float*)d_out;             // (32,512,512,1) f32
    probmask_undersample_kernel<<<GRID, BLOCK, 0, stream>>>(x, w, thr, out);
}